// GAT_89472758710577
// MI455X (gfx1250) — compile-verified
//
#include <hip/hip_runtime.h>
#include <math.h>

typedef float v2f __attribute__((ext_vector_type(2)));
typedef float v8f __attribute__((ext_vector_type(8)));

#define H 2
#define DIN 64
#define OUTD 32
#define LEAKY 0.2f

// ---------------------------------------------------------------------------
// zero-fill scratch (harness poisons d_ws with 0xAA)
// ---------------------------------------------------------------------------
__global__ void gat_zero(float* __restrict__ p, int n) {
    int i = blockIdx.x * blockDim.x + threadIdx.x;
    if (i < n) p[i] = 0.0f;
}

// ---------------------------------------------------------------------------
// wa[h,d] = sum_o W[h,d,o] * a[h,o]   (el = x . wa_l, er = x . wa_r)
// ---------------------------------------------------------------------------
__global__ void gat_wa(const float* __restrict__ W,
                       const float* __restrict__ al,
                       const float* __restrict__ ar,
                       float* __restrict__ wal, float* __restrict__ war) {
    int t = threadIdx.x;              // 128 threads: (h,d)
    int h = t >> 6, d = t & 63;
    float sl = 0.f, sr = 0.f;
#pragma unroll
    for (int o = 0; o < OUTD; ++o) {
        float w = W[h * (DIN * OUTD) + d * OUTD + o];
        sl += w * al[h * OUTD + o];
        sr += w * ar[h * OUTD + o];
    }
    wal[t] = sl;
    war[t] = sr;
}

// ---------------------------------------------------------------------------
// Z[N,64] = X[N,64] @ W'[64,64]  via V_WMMA_F32_16X16X4_F32.
// W'[d, h*32+o] = W[h,d,o].  Block = 256 thr = 8 waves; wave -> 16 rows x 64 cols
// (4 accumulator tiles).  Weights staged in LDS, K-PAIR INTERLEAVED so every
// B fragment is one contiguous ds_load_b64 into an even-aligned VGPR pair:
//   Wl[(kpair*64 + col)*2 + s] = W'[2*kpair + s][col],  s in {0,1}
// Lane layouts per CDNA5 ISA 7.12.2:
//   A 16x4 f32 : lanes 0-15 M=lane, VGPR{0,1}=K{0,1}; lanes 16-31 K{2,3}
//   B 4x16 f32 : lanes 0-15 N=lane, VGPR{0,1}=K{0,1}; lanes 16-31 K{2,3}
//   D 16x16 f32: VGPR i -> M = i + 8*(lane>=16), N = lane&15
// ---------------------------------------------------------------------------
__global__ void gat_proj_wmma(const float* __restrict__ X,
                              const float* __restrict__ W,   // [2,64,32]
                              float* __restrict__ Z, int N) {
    __shared__ float Wl[DIN * 64];                 // 16 KB, interleaved layout
    int tid = threadIdx.x;
    for (int i = tid; i < DIN * 64; i += blockDim.x) {
        int d = i >> 6, col = i & 63;              // source (d, col)
        int h = col >> 5, o = col & 31;
        int kpair = d >> 1, s = d & 1;
        Wl[(kpair * 64 + col) * 2 + s] = W[h * (DIN * OUTD) + d * OUTD + o];
    }
    __syncthreads();

    int wave = tid >> 5, lane = tid & 31;
    int half = lane >> 4, mrow = lane & 15;
    int r0 = blockIdx.x * 128 + wave * 16;
    if (r0 >= N) return;

    v8f acc0 = {}, acc1 = {}, acc2 = {}, acc3 = {};

    int arow = r0 + mrow;
    if (arow >= N) arow = N - 1;                   // clamped load; store predicated
    const v2f* xr = (const v2f*)(X + (long long)arow * DIN);
    const v2f* wp = (const v2f*)Wl;

#pragma unroll
    for (int kk = 0; kk < DIN; kk += 4) {
        int kp = (kk >> 1) + half;                 // K-pair index for this lane half
        v2f a  = xr[kp];                           // X[row][k0..k0+1], one b64 load
        v2f b0 = wp[kp * 64 + mrow];               // one ds_load_b64 each, even pair
        v2f b1 = wp[kp * 64 + 16 + mrow];
        v2f b2 = wp[kp * 64 + 32 + mrow];
        v2f b3 = wp[kp * 64 + 48 + mrow];

        acc0 = __builtin_amdgcn_wmma_f32_16x16x4_f32(false, a, false, b0, (short)0, acc0, false, false);
        acc1 = __builtin_amdgcn_wmma_f32_16x16x4_f32(false, a, false, b1, (short)0, acc1, false, false);
        acc2 = __builtin_amdgcn_wmma_f32_16x16x4_f32(false, a, false, b2, (short)0, acc2, false, false);
        acc3 = __builtin_amdgcn_wmma_f32_16x16x4_f32(false, a, false, b3, (short)0, acc3, false, false);
    }

#pragma unroll
    for (int i = 0; i < 8; ++i) {
        int row = r0 + i + 8 * half;
        if (row < N) {
            float* zr = Z + (long long)row * 64;
            zr[mrow]      = acc0[i];
            zr[16 + mrow] = acc1[i];
            zr[32 + mrow] = acc2[i];
            zr[48 + mrow] = acc3[i];
        }
    }
}

// ---------------------------------------------------------------------------
// scores[n*H+h] = x[n,:] . wa[h,:]
// ---------------------------------------------------------------------------
__global__ void gat_scores(const float* __restrict__ X,
                           const float* __restrict__ wa,
                           float* __restrict__ out, int n) {
    int idx = blockIdx.x * blockDim.x + threadIdx.x;
    if (idx >= n * H) return;
    int nd = idx >> 1, h = idx & 1;
    const float* x = X + (long long)nd * DIN;
    const float* w = wa + h * DIN;
    float s = 0.f;
#pragma unroll
    for (int d = 0; d < DIN; ++d) s += x[d] * w[d];
    out[idx] = s;
}

// order-preserving float->uint key (negatives -> [0,2^31), positives -> [2^31,2^32))
__device__ __forceinline__ unsigned fkey(float v) {
    unsigned u = __float_as_uint(v);
    return (u & 0x80000000u) ? ~u : (u | 0x80000000u);
}
__device__ __forceinline__ float fdec(unsigned k) {
    return (k & 0x80000000u) ? __uint_as_float(k & 0x7FFFFFFFu) : __uint_as_float(~k);
}

// e = leaky_relu(el[src]+er[dst]); segment max via uint atomicMax
__global__ void gat_edge_max(const int* __restrict__ src, const int* __restrict__ dst,
                             const float* __restrict__ el, const float* __restrict__ er,
                             float* __restrict__ ebuf, unsigned* __restrict__ mkey, int E) {
    int e = blockIdx.x * blockDim.x + threadIdx.x;
    if (e >= E) return;
    int s = src[e], d = dst[e];
#pragma unroll
    for (int h = 0; h < H; ++h) {
        float v = el[s * H + h] + er[d * H + h];
        v = v > 0.f ? v : LEAKY * v;
        ebuf[(long long)e * H + h] = v;
        atomicMax(&mkey[d * H + h], fkey(v));
    }
}

// w = exp(e - m[dst]); segment sum; overwrite ebuf with w
__global__ void gat_edge_sum(const int* __restrict__ dst,
                             float* __restrict__ ebuf,
                             const unsigned* __restrict__ mkey,
                             float* __restrict__ sbuf, int E) {
    int e = blockIdx.x * blockDim.x + threadIdx.x;
    if (e >= E) return;
    int d = dst[e];
#pragma unroll
    for (int h = 0; h < H; ++h) {
        float m = fdec(mkey[d * H + h]);
        float w = expf(ebuf[(long long)e * H + h] - m);
        ebuf[(long long)e * H + h] = w;
        atomicAdd(&sbuf[d * H + h], w);
    }
}

// acc[dst, c] += w * z[src, c]   (one thread per (edge, channel))
__global__ void gat_edge_accum(const int* __restrict__ src, const int* __restrict__ dst,
                               const float* __restrict__ ebuf,
                               const float* __restrict__ Z,
                               float* __restrict__ acc, long long EC) {
    long long idx = (long long)blockIdx.x * blockDim.x + threadIdx.x;
    if (idx >= EC) return;
    int e = (int)(idx >> 6);
    int c = (int)(idx & 63);
    int h = c >> 5;
    float w = ebuf[(long long)e * H + h];
    atomicAdd(&acc[(long long)dst[e] * 64 + c], w * Z[(long long)src[e] * 64 + c]);
}

// out = elu(acc / s)   (empty segments -> 0 -> elu(0)=0)
__global__ void gat_finalize(const float* __restrict__ acc,
                             const float* __restrict__ sbuf,
                             float* __restrict__ out, int n) {
    long long idx = (long long)blockIdx.x * blockDim.x + threadIdx.x;
    if (idx >= (long long)n * 64) return;
    int nd = (int)(idx >> 6);
    int c = (int)(idx & 63);
    int h = c >> 5;
    float sv = sbuf[nd * H + h];
    float r = sv > 0.f ? acc[idx] / sv : 0.f;
    out[idx] = r > 0.f ? r : expm1f(r);
}

// ---------------------------------------------------------------------------
// host orchestration
// ---------------------------------------------------------------------------
static void run_gat_layer(const float* x_src, int Ns, const float* x_dst, int Nd,
                          const float* W, const float* al, const float* ar,
                          const int* src, const int* dst, int E,
                          float* out, float* ws, hipStream_t stream) {
    // scratch carving (floats); sized for the larger (item->user) layer
    float*    Zb  = ws;                                  // 200000*64 = 12.8M
    float*    EB  = Zb + 12800000;                       //   2M*2    =  4.0M
    unsigned* MK  = (unsigned*)(EB + 4000000);           // 100000*2  =  0.2M
    float*    SB  = (float*)MK + 200000;                 //           =  0.2M
    float*    ACC = SB + 200000;                         // 100000*64 =  6.4M
    float*    EL  = ACC + 6400000;                       // 200000*2  =  0.4M
    float*    ER  = EL + 400000;                         // 100000*2  =  0.2M
    float*    WAL = ER + 200000;                         // 128
    float*    WAR = WAL + 128;                           // 128

    gat_wa<<<1, 128, 0, stream>>>(W, al, ar, WAL, WAR);
    gat_proj_wmma<<<(Ns + 127) / 128, 256, 0, stream>>>(x_src, W, Zb, Ns);
    gat_scores<<<(Ns * H + 255) / 256, 256, 0, stream>>>(x_src, WAL, EL, Ns);
    gat_scores<<<(Nd * H + 255) / 256, 256, 0, stream>>>(x_dst, WAR, ER, Nd);
    gat_zero<<<(Nd * H + 255) / 256, 256, 0, stream>>>((float*)MK, Nd * H);
    gat_zero<<<(Nd * H + 255) / 256, 256, 0, stream>>>(SB, Nd * H);
    gat_zero<<<(Nd * 64 + 255) / 256, 256, 0, stream>>>(ACC, Nd * 64);
    gat_edge_max<<<(E + 255) / 256, 256, 0, stream>>>(src, dst, EL, ER, EB, MK, E);
    gat_edge_sum<<<(E + 255) / 256, 256, 0, stream>>>(dst, EB, MK, SB, E);
    long long EC = (long long)E * 64;
    gat_edge_accum<<<(unsigned)((EC + 255) / 256), 256, 0, stream>>>(src, dst, EB, Zb, ACC, EC);
    gat_finalize<<<(unsigned)(((long long)Nd * 64 + 255) / 256), 256, 0, stream>>>(ACC, SB, out, Nd);
}

extern "C" void kernel_launch(void* const* d_in, const int* in_sizes, int n_in,
                              void* d_out, int out_size, void* d_ws, size_t ws_size,
                              hipStream_t stream) {
    const float* user_emb  = (const float*)d_in[0];
    const float* group_emb = (const float*)d_in[1];
    const float* item_emb  = (const float*)d_in[2];
    const float* W_iu  = (const float*)d_in[3];
    const float* al_iu = (const float*)d_in[4];
    const float* ar_iu = (const float*)d_in[5];
    const float* W_ug  = (const float*)d_in[6];
    const float* al_ug = (const float*)d_in[7];
    const float* ar_ug = (const float*)d_in[8];
    const int* src_iu = (const int*)d_in[9];
    const int* dst_iu = (const int*)d_in[10];
    const int* src_ug = (const int*)d_in[11];
    const int* dst_ug = (const int*)d_in[12];

    int n_users  = in_sizes[0] / DIN;
    int n_groups = in_sizes[1] / DIN;
    int n_items  = in_sizes[2] / DIN;
    int E_iu = in_sizes[9];
    int E_ug = in_sizes[11];

    float* out     = (float*)d_out;
    float* user_h  = out;                                // [n_users, 64]
    float* group_h = out + (long long)n_users * 64;      // [n_groups, 64]
    float* ws      = (float*)d_ws;

    // layer 0: item -> user, then elu (written into d_out, reused as layer-1 input)
    run_gat_layer(item_emb, n_items, user_emb, n_users,
                  W_iu, al_iu, ar_iu, src_iu, dst_iu, E_iu,
                  user_h, ws, stream);

    // layer 1: user -> group on updated user features, then elu
    run_gat_layer(user_h, n_users, group_emb, n_groups,
                  W_ug, al_ug, ar_ug, src_ug, dst_ug, E_ug,
                  group_h, ws, stream);
}